// GPUEventSimulator_5033701671352
// MI455X (gfx1250) — compile-verified
//
#include <hip/hip_runtime.h>

// Event-camera simulator for MI455X (gfx1250).
// Pure HBM-streaming workload: ~330 MB touched once -> ~14 us floor @ 23.3 TB/s.
// No GEMM structure => WMMA inapplicable; we instead exploit CDNA5 cache
// controls (non-temporal TH hints, global_prefetch_b8) and wave32 codegen.

#define BB 4
#define HH 480
#define WW 640
#define TT 32
#define HW (HH * WW)          // 307200
#define BHW (BB * HW)         // 1228800
#define BLOCK 256
#define BLOCKS_PER_B (HW / BLOCK)  // 1200

typedef float v4f __attribute__((ext_vector_type(4)));

__global__ __launch_bounds__(BLOCK)
void event_sim_kernel(const float* __restrict__ log_images,   // (H, W, B*T)
                      const float* __restrict__ image_ts,     // (B, T)
                      const float* __restrict__ first_times,  // (B,)
                      const float* __restrict__ thresholds,   // (2, B, H, W)
                      const float* __restrict__ log_states,   // (B, H, W)
                      const float* __restrict__ timestamps,   // (B, H, W)
                      const float* __restrict__ prev_image_ts,// (B,)
                      const float* __restrict__ refractory,   // (B,)
                      const float* __restrict__ leak_rates,   // (B,)
                      const float* __restrict__ shot_rates,   // (B,)
                      const float* __restrict__ threshold_mus,// (B, 2)
                      const float* __restrict__ rng,          // (T, B, H, W)
                      float* __restrict__ out)                // (B, H, W)
{
    // b is block-uniform -> per-batch params become SGPRs / s_loads.
    const int b   = blockIdx.x / BLOCKS_PER_B;
    const int hw  = (blockIdx.x % BLOCKS_PER_B) * BLOCK + threadIdx.x;
    const int tid = b * HW + hw;               // == rng per-t lane offset

    const float ref   = refractory[b];
    const float leak  = leak_rates[b];
    const float shot  = shot_rates[b];
    const float mu_on = threshold_mus[b * 2 + 1];
    const bool  ft    = first_times[b] > 0.0f;
    const float* __restrict__ ts_row = image_ts + b * TT;

    // All 32 frame samples for this pixel are one aligned 128B line:
    // flat = ((h*W + w)*B + b)*T + t. Load as 8x GLOBAL_LOAD_B128, TH=NT
    // (stream > 192MB L2; each byte read once).
    const v4f* __restrict__ fptr =
        (const v4f*)(log_images + ((size_t)hw * BB + (size_t)b) * TT);
    v4f f[8];
    #pragma unroll
    for (int i = 0; i < 8; ++i)
        f[i] = __builtin_nontemporal_load(fptr + i);

    const float th_off = __builtin_nontemporal_load(thresholds + tid);
    const float th_on  = __builtin_nontemporal_load(thresholds + BHW + tid);

    // Initial carry (reference's first_times handling).
    float state   = ft ? f[0][0]           : log_states[tid];
    float ts_prev = ft ? ts_row[0]         : prev_image_ts[b];
    float last    = ft ? (ts_prev - ref)   : timestamps[tid];
    float counts  = 0.0f;

    const float* __restrict__ rng_pix = rng + tid;

    #pragma unroll
    for (int t = 1; t < TT; ++t) {
        // gfx1250 global_prefetch_b8 for next timestep's rng cacheline.
        if (t + 1 < TT)
            __builtin_prefetch(rng_pix + (size_t)(t + 1) * BHW, 0, 0);

        const float ts_t  = ts_row[t];                                   // s_load
        const float u     = __builtin_nontemporal_load(
                                rng_pix + (size_t)t * BHW);              // coalesced NT
        const float frame = f[t >> 2][t & 3];

        const float dt = ts_t - ts_prev;
        state -= leak * dt * mu_on;

        const float diff = frame - state;
        const bool  pos  = diff >= 0.0f;
        const float th   = pos ? th_on : th_off;
        float n = floorf(fabsf(diff) / th);
        const float ok = (ts_t - last >= ref) ? 1.0f : 0.0f;
        n *= ok;
        const float pol = pos ? 1.0f : -1.0f;
        state  += pol * n * th;
        counts += pol * n;
        last    = (n > 0.0f) ? ts_t : last;

        // Shot noise event.
        const float p       = shot * dt;
        const bool  is_shot = (u < p) && (ts_t - last >= ref);
        const float pol_s   = (u < 0.5f * p) ? 1.0f : -1.0f;
        counts += is_shot ? pol_s : 0.0f;
        last    = is_shot ? ts_t : last;

        ts_prev = ts_t;
    }

    __builtin_nontemporal_store(counts, out + tid);
}

extern "C" void kernel_launch(void* const* d_in, const int* in_sizes, int n_in,
                              void* d_out, int out_size, void* d_ws, size_t ws_size,
                              hipStream_t stream) {
    (void)in_sizes; (void)n_in; (void)d_ws; (void)ws_size; (void)out_size;
    const float* log_images     = (const float*)d_in[0];
    // d_in[1] = video_len (int64) -- unused by the reference math.
    const float* image_ts       = (const float*)d_in[2];
    const float* first_times    = (const float*)d_in[3];
    const float* thresholds     = (const float*)d_in[4];
    const float* log_states     = (const float*)d_in[5];
    const float* timestamps     = (const float*)d_in[6];
    const float* prev_image_ts  = (const float*)d_in[7];
    const float* refractory     = (const float*)d_in[8];
    const float* leak_rates     = (const float*)d_in[9];
    const float* shot_rates     = (const float*)d_in[10];
    const float* threshold_mus  = (const float*)d_in[11];
    const float* rng            = (const float*)d_in[12];
    float* out                  = (float*)d_out;

    dim3 grid(BB * BLOCKS_PER_B);   // 4800 blocks
    dim3 block(BLOCK);              // 256 threads = 8 wave32 waves
    event_sim_kernel<<<grid, block, 0, stream>>>(
        log_images, image_ts, first_times, thresholds, log_states, timestamps,
        prev_image_ts, refractory, leak_rates, shot_rates, threshold_mus, rng, out);
}